// GATLayer_13194139533785
// MI455X (gfx1250) — compile-verified
//
#include <hip/hip_runtime.h>

typedef __attribute__((ext_vector_type(2))) float v2f;
typedef __attribute__((ext_vector_type(8))) float v8f;

#define NH 8
#define OD 32
#define NN 64
#define IN_DIM 256
#define HD 256          // NH*OD
#define ROWSTRIDE 260   // 256 + 4 floats pad: bank-conflict-free, keeps 16B align
#define VSTRIDE 260     // V^T rows padded the same way

// Async global->LDS staging (ASYNCcnt path), guarded so compile never breaks.
#if defined(__HIP_DEVICE_COMPILE__) && __has_builtin(__builtin_amdgcn_global_load_async_to_lds_b128)
#define USE_ASYNC_LDS 1
#else
#define USE_ASYNC_LDS 0
#endif

#if USE_ASYNC_LDS
typedef __attribute__((ext_vector_type(4))) int v4ia;
__device__ __forceinline__ void async_copy_b128(const float* gp, float* lp) {
    // builtin signature (from clang diagnostic): (v4i AS1*, v4i AS3*, imm offset, imm cpol)
    __builtin_amdgcn_global_load_async_to_lds_b128(
        (__attribute__((address_space(1))) v4ia*)(float*)gp,
        (__attribute__((address_space(3))) v4ia*)lp,
        0, 0);
}
#endif

// ---------------------------------------------------------------------------
// V^T[h][k] = sum_d W[k][h*32+d] * W_att[h][d]   (8 x 256, padded rows)
// ---------------------------------------------------------------------------
__global__ __launch_bounds__(256) void gat_prep_V(const float* __restrict__ W,
                                                  const float* __restrict__ W_att,
                                                  float* __restrict__ VT) {
    int k = threadIdx.x; // 0..255
    #pragma unroll
    for (int h = 0; h < NH; ++h) {
        float acc = 0.f;
        const float* wrow = W + k * HD + h * OD;
        const float* arow = W_att + h * OD;
        #pragma unroll
        for (int d = 0; d < OD; ++d) acc += wrow[d] * arow[d];
        VT[h * VSTRIDE + k] = acc;
    }
}

// ---------------------------------------------------------------------------
// One workgroup (8 wave32) per batch element b.
// ---------------------------------------------------------------------------
__global__ __launch_bounds__(256) void gat_main(const float* __restrict__ h_i,
                                                const float* __restrict__ h_j,
                                                const float* __restrict__ W,
                                                const float* __restrict__ VT,
                                                float* __restrict__ out) {
    __shared__ float hj_lds[NN * ROWSTRIDE];    // h_j rows           (65 KB)
    __shared__ float hi_lds[NN * ROWSTRIDE];    // h_i rows           (65 KB)
    __shared__ float v_lds [NH * VSTRIDE];      // V^T                ( 8 KB)
    __shared__ float alpha_lds[NN * 16];        // e / alpha, padded to 16 heads (4 KB)
    __shared__ float g_lds [NH * IN_DIM];       // G = alpha^T @ h_j  ( 8 KB)

    const int t = threadIdx.x;
    const int b = blockIdx.x;
    const float* hib = h_i + (size_t)b * NN * IN_DIM;
    const float* hjb = h_j + (size_t)b * NN * IN_DIM;

    // zero alpha padding (heads 8..15 must be 0 for the WMMA A-operand)
    #pragma unroll
    for (int i = 0; i < 4; ++i) alpha_lds[t + i * 256] = 0.f;

    // stage V^T (small, regular path)
    for (int i = t; i < NH * VSTRIDE; i += 256) v_lds[i] = VT[i];

    // ---- stage h_i, h_j (64 KB each) ----
#if USE_ASYNC_LDS
    #pragma unroll
    for (int j = 0; j < 16; ++j) {
        int f   = t + j * 256;   // float4 index 0..4095
        int row = f >> 6;
        int c4  = f & 63;
        async_copy_b128(hjb + f * 4, &hj_lds[row * ROWSTRIDE + c4 * 4]);
        async_copy_b128(hib + f * 4, &hi_lds[row * ROWSTRIDE + c4 * 4]);
    }
    #if __has_builtin(__builtin_amdgcn_s_wait_asynccnt)
    __builtin_amdgcn_s_wait_asynccnt(0);
    #else
    asm volatile("s_wait_asynccnt 0" ::: "memory");
    #endif
#else
    #pragma unroll
    for (int j = 0; j < 16; ++j) {
        int f   = t + j * 256;
        int row = f >> 6;
        int c4  = f & 63;
        *(float4*)&hj_lds[row * ROWSTRIDE + c4 * 4] = ((const float4*)hjb)[f];
        *(float4*)&hi_lds[row * ROWSTRIDE + c4 * 4] = ((const float4*)hib)[f];
    }
#endif
    __syncthreads();

    // e[n][h] = leakyReLU( (h_i+h_j)[n,:] @ V[:,h], 0.2 )  -> alpha_lds
    {
        int n  = t >> 2;
        int h2 = (t & 3) * 2;
        const float4* hi4 = (const float4*)&hi_lds[n * ROWSTRIDE];
        const float4* hj4 = (const float4*)&hj_lds[n * ROWSTRIDE];
        const float4* va4 = (const float4*)&v_lds[h2 * VSTRIDE];
        const float4* vb4 = (const float4*)&v_lds[(h2 + 1) * VSTRIDE];
        float a0 = 0.f, a1 = 0.f;
        #pragma unroll 8
        for (int q = 0; q < IN_DIM / 4; ++q) {
            float4 x = hi4[q], y = hj4[q];
            float4 va = va4[q], vb = vb4[q];
            float s0 = x.x + y.x, s1 = x.y + y.y, s2 = x.z + y.z, s3 = x.w + y.w;
            a0 += s0 * va.x + s1 * va.y + s2 * va.z + s3 * va.w;
            a1 += s0 * vb.x + s1 * vb.y + s2 * vb.z + s3 * vb.w;
        }
        a0 = a0 > 0.f ? a0 : 0.2f * a0;
        a1 = a1 > 0.f ? a1 : 0.2f * a1;
        alpha_lds[n * 16 + h2]     = a0;
        alpha_lds[n * 16 + h2 + 1] = a1;
    }
    __syncthreads();

    // softmax over n (64) per head, in place (8 lanes, tiny)
    if (t < NH) {
        float m = -3.4e38f;
        for (int n = 0; n < NN; ++n) m = fmaxf(m, alpha_lds[n * 16 + t]);
        float ssum = 0.f;
        for (int n = 0; n < NN; ++n) {
            float ev = __expf(alpha_lds[n * 16 + t] - m);
            alpha_lds[n * 16 + t] = ev;
            ssum += ev;
        }
        float inv = 1.f / ssum;
        for (int n = 0; n < NN; ++n) alpha_lds[n * 16 + t] *= inv;
    }
    __syncthreads();

    // G[16,256] = alpha^T[16,64] @ h_j[64,256]  via V_WMMA_F32_16X16X4_F32
    // A 16x4 layout: VGPR0 = K{0 | 2}, VGPR1 = K{1 | 3} across lane halves; B 4x16 mirrors it.
    {
        int lane  = t & 31;
        int w     = t >> 5;
        int m     = lane & 15;
        int khalf = (lane >> 4) * 2;
        v8f acc0 = {0.f, 0.f, 0.f, 0.f, 0.f, 0.f, 0.f, 0.f};
        v8f acc1 = acc0;
        const int nt0 = w * 2, nt1 = w * 2 + 1;   // 16 N-tiles over 8 waves
        const int c0 = nt0 * 16 + m, c1 = nt1 * 16 + m;
        for (int kk = 0; kk < 16; ++kk) {
            int k0 = kk * 4 + khalf;
            v2f a, b0, b1;
            a.x  = alpha_lds[k0 * 16 + m];
            a.y  = alpha_lds[(k0 + 1) * 16 + m];
            b0.x = hj_lds[k0 * ROWSTRIDE + c0];
            b0.y = hj_lds[(k0 + 1) * ROWSTRIDE + c0];
            b1.x = hj_lds[k0 * ROWSTRIDE + c1];
            b1.y = hj_lds[(k0 + 1) * ROWSTRIDE + c1];
            acc0 = __builtin_amdgcn_wmma_f32_16x16x4_f32(false, a, false, b0,
                                                         (short)0, acc0, false, false);
            acc1 = __builtin_amdgcn_wmma_f32_16x16x4_f32(false, a, false, b1,
                                                         (short)0, acc1, false, false);
        }
        // D rows 0..7 (real heads) live in lanes 0..15 of acc[r]
        if (lane < 16) {
            #pragma unroll
            for (int r = 0; r < NH; ++r) {
                g_lds[r * IN_DIM + nt0 * 16 + m] = acc0[r];
                g_lds[r * IN_DIM + nt1 * 16 + m] = acc1[r];
            }
        }
    }
    __syncthreads();

    // out[b, h*32+d] = elu( G[h,:] @ W[:, h*32+d] )   (W stays L2-resident, 256 KB)
    {
        int h = t >> 5;
        int d = t & 31;
        float acc = 0.f;
        #pragma unroll 4
        for (int k = 0; k < IN_DIM; ++k)
            acc += g_lds[h * IN_DIM + k] * W[k * HD + h * OD + d];
        float r = acc > 0.f ? acc : (__expf(acc) - 1.f);
        out[(size_t)b * HD + h * OD + d] = r;
    }
}

extern "C" void kernel_launch(void* const* d_in, const int* in_sizes, int n_in,
                              void* d_out, int out_size, void* d_ws, size_t ws_size,
                              hipStream_t stream) {
    const float* h_i   = (const float*)d_in[0];
    const float* h_j   = (const float*)d_in[1];
    const float* W     = (const float*)d_in[2];
    const float* W_att = (const float*)d_in[3];
    float* out = (float*)d_out;
    float* VT  = (float*)d_ws;               // 8*260 floats ≈ 8.1 KB scratch

    const int B = in_sizes[0] / (NN * IN_DIM);

    gat_prep_V<<<1, 256, 0, stream>>>(W, W_att, VT);
    gat_main<<<B, 256, 0, stream>>>(h_i, h_j, W, VT, out);
}